// DepthPoseLosses_12824772346462
// MI455X (gfx1250) — compile-verified
//
#include <hip/hip_runtime.h>

typedef float v2f __attribute__((ext_vector_type(2)));
typedef float v8f __attribute__((ext_vector_type(8)));

#define B_ 8
#define H0 256
#define W0 832
#define SSIM_C1 1e-4f
#define SSIM_C2 9e-4f

// ---------------- workspace layout (float offsets) ----------------
constexpr int RS1 = B_ * 3 * (H0 / 2) * (W0 / 2);
constexpr int RS2 = B_ * 3 * (H0 / 4) * (W0 / 4);
constexpr int RS3 = B_ * 3 * (H0 / 8) * (W0 / 8);
constexpr size_t OFF_ACC   = 0;    // [0..3]=DP,DC,PC,DS  [4..8]=per-invocation sums
constexpr size_t OFF_MEAND = 16;   // 4 scales * 3 imgs * 8 batches
constexpr size_t OFF_POSE  = 112;  // 3 pairs * 2 dirs * 8 batches * 12 (P0 = K@[R|t])
constexpr size_t OFF_RES1  = OFF_POSE + 576;
constexpr size_t OFF_RES2  = OFF_RES1 + 3 * (size_t)RS1;
constexpr size_t OFF_RES3  = OFF_RES2 + 3 * (size_t)RS2;
constexpr size_t OFF_WARP  = OFF_RES3 + 3 * (size_t)RS3;
constexpr size_t OFF_DCONS = OFF_WARP + (size_t)B_ * 3 * H0 * W0;
constexpr size_t OFF_VALID = OFF_DCONS + (size_t)B_ * H0 * W0;

// ---------------- helpers ----------------
__device__ __forceinline__ float blockReduceSum(float v) {
  __shared__ float red[8];
  int t = threadIdx.y * blockDim.x + threadIdx.x;
  int lane = t & 31, wid = t >> 5;
#pragma unroll
  for (int o = 16; o > 0; o >>= 1) v += __shfl_down(v, o, 32);
  __syncthreads();
  if (lane == 0) red[wid] = v;
  __syncthreads();
  float r = 0.f;
  if (wid == 0) {
    r = (lane < 8) ? red[lane] : 0.f;
#pragma unroll
    for (int o = 4; o > 0; o >>= 1) r += __shfl_down(r, o, 32);
  }
  return r;  // valid on linear tid 0
}

// ---------------- kernels ----------------
__global__ void k_zero(float* p, int n) {
  int i = blockIdx.x * blockDim.x + threadIdx.x;
  if (i < n) p[i] = 0.f;
}

// bilinear resize (jax.image.resize, antialias=False), always from full-res
__global__ __launch_bounds__(256) void k_resize(const float* __restrict__ src,
                                                float* __restrict__ dst,
                                                int h, int w, int sc) {
  int idx = blockIdx.x * blockDim.x + threadIdx.x;
  int n = B_ * 3 * h * w;
  if (idx >= n) return;
  int x = idx % w;
  int t = idx / w;
  int y = t % h;
  int bc = t / h;
  float s = (float)sc;
  float sy = (y + 0.5f) * s - 0.5f;
  float sx = (x + 0.5f) * s - 0.5f;
  int y0 = (int)floorf(sy), x0 = (int)floorf(sx);
  float wy = sy - (float)y0, wx = sx - (float)x0;
  int y1 = min(y0 + 1, H0 - 1), x1 = min(x0 + 1, W0 - 1);
  y0 = max(y0, 0); x0 = max(x0, 0);
  const float* p = src + (size_t)bc * H0 * W0;
  float v00 = p[y0 * W0 + x0], v01 = p[y0 * W0 + x1];
  float v10 = p[y1 * W0 + x0], v11 = p[y1 * W0 + x1];
  dst[idx] = v00 * (1.f - wy) * (1.f - wx) + v01 * (1.f - wy) * wx +
             v10 * wy * (1.f - wx) + v11 * wy * wx;
}

// per (img,batch) mean of a depth plane; one block per plane
__global__ __launch_bounds__(256) void k_meand(const float* __restrict__ d,
                                               float* __restrict__ out, int n) {
  int blk = blockIdx.x;
  const float* p = d + (size_t)blk * n;
  float s = 0.f;
  for (int i = threadIdx.x; i < n; i += blockDim.x) s += p[i];
  s = blockReduceSum(s);
  if (threadIdx.x == 0) out[blk] = s / (float)n;
}

// Pose kernel: builds 4x4 pose matrices, computes P0 = K @ [R|t] and the
// pose-consistency loss via V_WMMA_F32_16X16X4_F32 (4 batched 4x4 products
// per WMMA, packed block-diagonally). One wave per pair => EXEC all ones.
__global__ __launch_bounds__(32) void k_pose(const float* __restrict__ poses,
                                             const float* __restrict__ poses_inv,
                                             const float* __restrict__ Kmat,
                                             float* __restrict__ P0out,
                                             float* __restrict__ acc) {
  int pair = blockIdx.x;
  int lane = threadIdx.x;
  __shared__ float Ma[8][16];
  __shared__ float Mb[8][16];
  if (lane < 16) {
    int b = lane & 7;
    const float* v = ((lane < 8) ? poses : poses_inv) + ((size_t)pair * 8 + b) * 6;
    float tx = v[0], ty = v[1], tz = v[2];
    float ax = v[3], ay = v[4], az = v[5];
    float cz = cosf(az), sz = sinf(az);
    float cy = cosf(ay), sy = sinf(ay);
    float cx = cosf(ax), sx = sinf(ax);
    float M[16];
    M[0] = cy * cz;                 M[1] = -cy * sz;                M[2] = sy;        M[3] = tx;
    M[4] = cx * sz + sx * sy * cz;  M[5] = cx * cz - sx * sy * sz;  M[6] = -sx * cy;  M[7] = ty;
    M[8] = sx * sz - cx * sy * cz;  M[9] = sx * cz + cx * sy * sz;  M[10] = cx * cy;  M[11] = tz;
    M[12] = 0.f; M[13] = 0.f; M[14] = 0.f; M[15] = 1.f;
    float(*dst)[16] = (lane < 8) ? Ma : Mb;
#pragma unroll
    for (int k = 0; k < 16; k++) dst[b][k] = M[k];
    const float* Kb = Kmat + (size_t)b * 9;
    int dir = (lane < 8) ? 0 : 1;
    float* P = P0out + (((size_t)pair * 2 + dir) * 8 + b) * 12;
#pragma unroll
    for (int r = 0; r < 3; r++)
#pragma unroll
      for (int c = 0; c < 4; c++)
        P[r * 4 + c] = Kb[r * 3 + 0] * M[0 + c] + Kb[r * 3 + 1] * M[4 + c] +
                       Kb[r * 3 + 2] * M[8 + c];
  }
  __syncthreads();
  int m = lane & 15;          // row (A) / col (B/D) index
  int hi = lane >> 4;         // lane half selects K rows {0,1} vs {2,3}
  int bi = m >> 2;            // batch within group of 4
  int r = m & 3;              // row/col within 4x4 block
  float pcsum = 0.f;
#pragma unroll
  for (int ord = 0; ord < 2; ++ord) {
#pragma unroll
    for (int g = 0; g < 2; ++g) {
      int ab = g * 4 + bi;
      float(*L)[16] = ord ? Mb : Ma;
      float(*R)[16] = ord ? Ma : Mb;
      v2f A, Bf;
      // A (16x4): A[m][k] = L[ab][r*4+k]; lanes 0-15 hold K=0,1 ; 16-31 hold K=2,3
      A.x = L[ab][r * 4 + (hi ? 2 : 0)];
      A.y = L[ab][r * 4 + (hi ? 3 : 1)];
      // B (4x16): B[k][n] = R[ab][k*4 + (n&3)]; VGPR0 rows {0,2}, VGPR1 rows {1,3}
      Bf.x = R[ab][(hi ? 2 : 0) * 4 + r];
      Bf.y = R[ab][(hi ? 3 : 1) * 4 + r];
      v8f C = {};
      v8f D = __builtin_amdgcn_wmma_f32_16x16x4_f32(false, A, false, Bf, (short)0,
                                                    C, false, false);
#pragma unroll
      for (int j = 0; j < 8; j++) {
        int MM = j + (hi ? 8 : 0);
        int NN = m;
        if ((MM >> 2) == (NN >> 2)) {  // diagonal 4x4 block = per-batch product
          float tgt = (MM == NN) ? 1.f : 0.f;
          pcsum += fabsf(D[j] - tgt);
        }
      }
    }
  }
  for (int o = 16; o > 0; o >>= 1) pcsum += __shfl_down(pcsum, o, 32);
  if (lane == 0) atomicAdd(&acc[2], 4.f * pcsum / 128.f);  // 4 scales x (pc1+pc2)
}

// Inverse warp: back-project, transform, project, bilinear-gather ref image
// and ref depth, depth-consistency ratio, validity mask.
__global__ __launch_bounds__(256) void k_warp(
    const float* __restrict__ tgt_d, const float* __restrict__ ref_d,
    const float* __restrict__ ref_img, const float* __restrict__ P0,
    const float* __restrict__ Kmat, float* __restrict__ warped,
    float* __restrict__ dcons, float* __restrict__ valid, int h, int w,
    float dsf) {
  int idx = blockIdx.x * blockDim.x + threadIdx.x;
  int hw = h * w;
  int n = B_ * hw;
  if (idx >= n) return;
  int b = idx / hw;
  int rem = idx - b * hw;
  int y = rem / w;
  int x = rem - y * w;
  const float* Kb = Kmat + (size_t)b * 9;
  float fx = Kb[0], cxk = Kb[2], fy = Kb[4], cyk = Kb[5];
  float d = tgt_d[idx];
  // cam = inv(K_s) @ [x,y,1] * d   (analytic; K_s rows 0-1 = K/ds)
  float camx = d * (dsf * (float)x - cxk) / fx;
  float camy = d * (dsf * (float)y - cyk) / fy;
  float camz = d;
  const float* P = P0 + (size_t)b * 12;  // P0 = K @ [R|t] at scale 0
  float pcx = (P[0] * camx + P[1] * camy + P[2] * camz + P[3]) / dsf;
  float pcy = (P[4] * camx + P[5] * camy + P[6] * camz + P[7]) / dsf;
  float pcz = P[8] * camx + P[9] * camy + P[10] * camz + P[11];
  float Z = fmaxf(pcz, 0.001f);
  float Xn = 2.f * (pcx / Z) / (float)(w - 1) - 1.f;
  float Yn = 2.f * (pcy / Z) / (float)(h - 1) - 1.f;
  valid[idx] = (fmaxf(fabsf(Xn), fabsf(Yn)) <= 1.f) ? 1.f : 0.f;
  float ix = (Xn + 1.f) * 0.5f * (float)(w - 1);
  float iy = (Yn + 1.f) * 0.5f * (float)(h - 1);
  float fx0 = floorf(ix), fy0 = floorf(iy);
  float wxf = ix - fx0, wyf = iy - fy0;
  float fx1 = fx0 + 1.f, fy1 = fy0 + 1.f;
  float fw = (float)w, fh = (float)h;
  float in00 = (fx0 >= 0.f && fx0 < fw && fy0 >= 0.f && fy0 < fh) ? 1.f : 0.f;
  float in01 = (fx1 >= 0.f && fx1 < fw && fy0 >= 0.f && fy0 < fh) ? 1.f : 0.f;
  float in10 = (fx0 >= 0.f && fx0 < fw && fy1 >= 0.f && fy1 < fh) ? 1.f : 0.f;
  float in11 = (fx1 >= 0.f && fx1 < fw && fy1 >= 0.f && fy1 < fh) ? 1.f : 0.f;
  int xi0 = (int)fminf(fmaxf(fx0, 0.f), fw - 1.f);
  int xi1 = (int)fminf(fmaxf(fx1, 0.f), fw - 1.f);
  int yi0 = (int)fminf(fmaxf(fy0, 0.f), fh - 1.f);
  int yi1 = (int)fminf(fmaxf(fy1, 0.f), fh - 1.f);
  float w00 = (1.f - wyf) * (1.f - wxf) * in00;
  float w01 = (1.f - wyf) * wxf * in01;
  float w10 = wyf * (1.f - wxf) * in10;
  float w11 = wyf * wxf * in11;
#pragma unroll
  for (int c = 0; c < 3; c++) {
    const float* base = ref_img + ((size_t)b * 3 + c) * hw;
    float val = base[yi0 * w + xi0] * w00 + base[yi0 * w + xi1] * w01 +
                base[yi1 * w + xi0] * w10 + base[yi1 * w + xi1] * w11;
    warped[((size_t)b * 3 + c) * hw + rem] = val;
  }
  const float* dbase = ref_d + (size_t)b * hw;
  float pd = dbase[yi0 * w + xi0] * w00 + dbase[yi0 * w + xi1] * w01 +
             dbase[yi1 * w + xi0] * w10 + dbase[yi1 * w + xi1] * w11;
  pd = fmaxf(pd, 0.001f);
  dcons[idx] = fabsf(Z - pd) / fabsf(Z + pd);
}

// Fused SSIM (3x3 reflect box via LDS halo tiles) + photometric + auto-mask +
// smoothness partials; block-reduced partial sums -> atomicAdd into workspace.
__global__ __launch_bounds__(256) void k_ssim(
    const float* __restrict__ tgt, const float* __restrict__ ref,
    const float* __restrict__ tgt_d, const float* __restrict__ warped,
    const float* __restrict__ dcons, const float* __restrict__ valid,
    const float* __restrict__ meand, float* __restrict__ accS, int h, int w) {
  __shared__ float sT[18 * 19];
  __shared__ float sW[18 * 19];
  int tid = threadIdx.y * 16 + threadIdx.x;
  int b = blockIdx.z;
  int x = blockIdx.x * 16 + threadIdx.x;
  int y = blockIdx.y * 16 + threadIdx.y;
  bool act = (x < w) && (y < h);
  size_t hw = (size_t)h * w;
  float photoSum = 0.f, aN = 0.f, aD = 0.f, gix = 0.f, giy = 0.f;
  for (int c = 0; c < 3; c++) {
    const float* tp = tgt + ((size_t)b * 3 + c) * hw;
    const float* wp = warped + ((size_t)b * 3 + c) * hw;
    __syncthreads();
    for (int i = tid; i < 18 * 18; i += 256) {
      int rr = i / 18, qq = i % 18;
      int gy = blockIdx.y * 16 - 1 + rr;
      int gx = blockIdx.x * 16 - 1 + qq;
      if (gy < 0) gy = -gy;
      if (gy >= h) gy = 2 * h - 2 - gy;
      if (gx < 0) gx = -gx;
      if (gx >= w) gx = 2 * w - 2 - gx;
      sT[rr * 19 + qq] = tp[(size_t)gy * w + gx];
      sW[rr * 19 + qq] = wp[(size_t)gy * w + gx];
    }
    __syncthreads();
    if (act) {
      int lx = threadIdx.x, ly = threadIdx.y;
      float s1 = 0.f, s2 = 0.f, s11 = 0.f, s22 = 0.f, s12 = 0.f;
#pragma unroll
      for (int dy = 0; dy < 3; dy++)
#pragma unroll
        for (int dx = 0; dx < 3; dx++) {
          float a = sT[(ly + dy) * 19 + lx + dx];
          float bb = sW[(ly + dy) * 19 + lx + dx];
          s1 += a; s2 += bb; s11 += a * a; s22 += bb * bb; s12 += a * bb;
        }
      float mux = s1 / 9.f, muy = s2 / 9.f;
      float sigx = s11 / 9.f - mux * mux;
      float sigy = s22 / 9.f - muy * muy;
      float sigxy = s12 / 9.f - mux * muy;
      float nn = (2.f * mux * muy + SSIM_C1) * (2.f * sigxy + SSIM_C2);
      float dd = (mux * mux + muy * muy + SSIM_C1) * (sigx + sigy + SSIM_C2);
      float ss = (1.f - nn / dd) * 0.5f;
      ss = fminf(fmaxf(ss, 0.f), 1.f);
      float xc = sT[(ly + 1) * 19 + lx + 1];
      float yc = sW[(ly + 1) * 19 + lx + 1];
      float l1 = fabsf(xc - yc);
      photoSum += 0.85f * ss + 0.15f * fminf(l1, 1.f);
      aN += l1;
      aD += fabsf(xc - ref[((size_t)b * 3 + c) * hw + (size_t)y * w + x]);
      if (x + 1 < w) gix += fabsf(xc - sT[(ly + 1) * 19 + lx + 2]);
      if (y + 1 < h) giy += fabsf(xc - sT[(ly + 2) * 19 + lx + 1]);
    }
  }
  float pm = 0.f, pdc = 0.f, pph = 0.f, pgx = 0.f, pgy = 0.f;
  if (act) {
    size_t pix = (size_t)b * hw + (size_t)y * w + x;
    float vr = valid[pix];
    float dc = dcons[pix];
    float am = ((aN / 3.f) < (aD / 3.f)) ? 1.f : 0.f;
    float vm = am * vr;
    pm = vm;
    pdc = dc * vm;
    pph = (photoSum / 3.f) * (1.f - dc) * vm;
    float md = meand[b] + 1e-7f;
    float d00 = tgt_d[pix];
    if (x + 1 < w) pgx = (fabsf(d00 - tgt_d[pix + 1]) / md) * expf(-gix / 3.f);
    if (y + 1 < h) pgy = (fabsf(d00 - tgt_d[pix + w]) / md) * expf(-giy / 3.f);
  }
  float r0 = blockReduceSum(pdc);
  float r1 = blockReduceSum(pm);
  float r2 = blockReduceSum(pph);
  float r3 = blockReduceSum(pgx);
  float r4 = blockReduceSum(pgy);
  if (tid == 0) {
    atomicAdd(&accS[0], r0);
    atomicAdd(&accS[1], r1);
    atomicAdd(&accS[2], r2);
    atomicAdd(&accS[3], r3);
    atomicAdd(&accS[4], r4);
  }
}

// Per-invocation finalize: apply mask>100 conditional, normalize smoothness.
__global__ void k_final_inv(float* acc, int h, int w) {
  float sd = acc[4], sm = acc[5], sp = acc[6], sgx = acc[7], sgy = acc[8];
  float den = fmaxf(sm, 1.f);
  float dc = (sm > 100.f) ? sd / den : 0.f;
  float ph = (sm > 100.f) ? sp / den : 0.f;
  float dsm = sgx / (float)(B_ * h * (w - 1)) + sgy / (float)(B_ * (h - 1) * w);
  acc[0] += ph;
  acc[1] += dc;
  acc[3] += dsm;
  acc[4] = 0.f; acc[5] = 0.f; acc[6] = 0.f; acc[7] = 0.f; acc[8] = 0.f;
}

__global__ void k_out(const float* acc, float* out) {
  int i = threadIdx.x;
  if (i < 4) out[i] = acc[i] / 3.f;
}

// ---------------- host driver ----------------
extern "C" void kernel_launch(void* const* d_in, const int* in_sizes, int n_in,
                              void* d_out, int out_size, void* d_ws,
                              size_t ws_size, hipStream_t stream) {
  (void)in_sizes; (void)n_in; (void)out_size; (void)ws_size;
  const float* imgs = (const float*)d_in[0];
  const float* depths[4] = {(const float*)d_in[1], (const float*)d_in[2],
                            (const float*)d_in[3], (const float*)d_in[4]};
  const float* poses = (const float*)d_in[5];
  const float* poses_inv = (const float*)d_in[6];
  const float* Kmat = (const float*)d_in[7];
  float* ws = (float*)d_ws;
  float* out = (float*)d_out;

  k_zero<<<1, 32, 0, stream>>>(ws + OFF_ACC, 16);

  const size_t resOff[4] = {0, OFF_RES1, OFF_RES2, OFF_RES3};
  const size_t rsz[4] = {0, (size_t)RS1, (size_t)RS2, (size_t)RS3};
  for (int s = 1; s < 4; s++) {
    int h = H0 >> s, w = W0 >> s;
    int n = B_ * 3 * h * w;
    for (int a = 0; a < 3; a++) {
      k_resize<<<(n + 255) / 256, 256, 0, stream>>>(
          imgs + (size_t)a * B_ * 3 * H0 * W0, ws + resOff[s] + (size_t)a * rsz[s],
          h, w, 1 << s);
    }
  }
  for (int s = 0; s < 4; s++) {
    int h = H0 >> s, w = W0 >> s;
    k_meand<<<24, 256, 0, stream>>>(depths[s], ws + OFF_MEAND + s * 24, h * w);
  }
  k_pose<<<3, 32, 0, stream>>>(poses, poses_inv, Kmat, ws + OFF_POSE, ws + OFF_ACC);

  const int PA[3] = {0, 0, 1}, PB[3] = {1, 2, 2};
  const size_t imgStride = (size_t)B_ * 3 * H0 * W0;
  for (int i = 0; i < 3; i++) {
    for (int s = 0; s < 4; s++) {
      int h = H0 >> s, w = W0 >> s;
      size_t hw = (size_t)h * w;
      int n = B_ * h * w;
      for (int dir = 0; dir < 2; dir++) {
        int ta = dir ? PB[i] : PA[i];
        int ra = dir ? PA[i] : PB[i];
        const float* tgtI = (s == 0) ? imgs + (size_t)ta * imgStride
                                     : ws + resOff[s] + (size_t)ta * rsz[s];
        const float* refI = (s == 0) ? imgs + (size_t)ra * imgStride
                                     : ws + resOff[s] + (size_t)ra * rsz[s];
        const float* tgtD = depths[s] + (size_t)ta * B_ * hw;
        const float* refD = depths[s] + (size_t)ra * B_ * hw;
        const float* P0 = ws + OFF_POSE + ((size_t)i * 2 + dir) * 8 * 12;
        k_warp<<<(n + 255) / 256, 256, 0, stream>>>(
            tgtD, refD, refI, P0, Kmat, ws + OFF_WARP, ws + OFF_DCONS,
            ws + OFF_VALID, h, w, (float)(1 << s));
        dim3 g((w + 15) / 16, (h + 15) / 16, B_);
        k_ssim<<<g, dim3(16, 16), 0, stream>>>(
            tgtI, refI, tgtD, ws + OFF_WARP, ws + OFF_DCONS, ws + OFF_VALID,
            ws + OFF_MEAND + s * 24 + ta * 8, ws + OFF_ACC + 4, h, w);
        k_final_inv<<<1, 1, 0, stream>>>(ws + OFF_ACC, h, w);
      }
    }
  }
  k_out<<<1, 32, 0, stream>>>(ws + OFF_ACC, out);
}